// PreTrainingModel_20633022890414
// MI455X (gfx1250) — compile-verified
//
#include <hip/hip_runtime.h>

// LSS / nuScenes configuration (must match reference)
#define BB 8
#define NN 6
#define DDZ 41
#define FHH 16
#define FWW 44
#define CC 80
#define NXV 200
#define NYV 200
#define NZV 1
#define NPTS (BB*NN*DDZ*FHH*FWW)      // 1,385,472
#define PTS_PER_BN (DDZ*FHH*FWW)      // 28,864 (divisible by 16)
#define NGROUPS (NPTS/16)             // 86,592 (divisible by 8)
#define PLANE (NYV*NXV)               // 40,000

typedef __attribute__((ext_vector_type(2))) float v2f;
typedef __attribute__((ext_vector_type(4))) float v4f;
typedef __attribute__((ext_vector_type(8))) float v8f;

__device__ inline void inv3x3d(const float* m, double inv[9]) {
    double a00=m[0],a01=m[1],a02=m[2],a10=m[3],a11=m[4],a12=m[5],a20=m[6],a21=m[7],a22=m[8];
    double det = a00*(a11*a22-a12*a21) - a01*(a10*a22-a12*a20) + a02*(a10*a21-a11*a20);
    double id = 1.0/det;
    inv[0]=(a11*a22-a12*a21)*id; inv[1]=(a02*a21-a01*a22)*id; inv[2]=(a01*a12-a02*a11)*id;
    inv[3]=(a12*a20-a10*a22)*id; inv[4]=(a00*a22-a02*a20)*id; inv[5]=(a02*a10-a00*a12)*id;
    inv[6]=(a10*a21-a11*a20)*id; inv[7]=(a01*a20-a00*a21)*id; inv[8]=(a00*a11-a01*a10)*id;
}

// Kernel 1: build per-(b,n) fused affines, pre-expanded into the exact
// per-lane A-matrix register images for V_WMMA_F32_16X16X4_F32.
// Layout per bn: wsMat[bn*128 + mat*64 + vgpr*32 + lane]
//   mat 0: [invPR | -invPR*post_trans],  mat 1: [rot*invK | trans]
// A-matrix (16x4) lane layout: lane L<16: (K0,K1) of row L; lane L>=16: (K2,K3) of row L-16.
__global__ void setup_mats(const float* __restrict__ rots, const float* __restrict__ trans,
                           const float* __restrict__ intrins, const float* __restrict__ post_rots,
                           const float* __restrict__ post_trans, float* __restrict__ wsMat) {
    int bn = blockIdx.x * blockDim.x + threadIdx.x;
    if (bn >= BB*NN) return;
    double iPR[9], iK[9];
    inv3x3d(post_rots + bn*9, iPR);
    inv3x3d(intrins  + bn*9, iK);
    const float* ptr = post_trans + bn*3;
    double a[3];
    for (int i=0;i<3;i++)
        a[i] = -(iPR[i*3+0]*(double)ptr[0] + iPR[i*3+1]*(double)ptr[1] + iPR[i*3+2]*(double)ptr[2]);
    const float* R = rots + bn*9;
    double M[9];
    for (int i=0;i<3;i++)
        for (int j=0;j<3;j++)
            M[i*3+j] = (double)R[i*3+0]*iK[0*3+j] + (double)R[i*3+1]*iK[1*3+j] + (double)R[i*3+2]*iK[2*3+j];
    const float* tr = trans + bn*3;

    float X[2][12];   // [mat][row*4 + col], col 3 = translation
    for (int i=0;i<3;i++) {
        X[0][i*4+0]=(float)iPR[i*3+0]; X[0][i*4+1]=(float)iPR[i*3+1];
        X[0][i*4+2]=(float)iPR[i*3+2]; X[0][i*4+3]=(float)a[i];
        X[1][i*4+0]=(float)M[i*3+0];   X[1][i*4+1]=(float)M[i*3+1];
        X[1][i*4+2]=(float)M[i*3+2];   X[1][i*4+3]=tr[i];
    }
    float* base = wsMat + bn*128;
    for (int mat=0; mat<2; mat++) {
        for (int lane=0; lane<32; lane++) {
            int Lr = lane & 15;
            float v0 = 0.f, v1 = 0.f;
            if (Lr < 3) {
                if (lane < 16) { v0 = X[mat][Lr*4+0]; v1 = X[mat][Lr*4+1]; }
                else           { v0 = X[mat][Lr*4+2]; v1 = X[mat][Lr*4+3]; }
            }
            base[mat*64 + 0*32 + lane] = v0;
            base[mat*64 + 1*32 + lane] = v1;
        }
    }
}

// Kernel 2: geometry via two chained V_WMMA_F32_16X16X4_F32 ops.
// One wave processes 16 points per iteration (both lane halves carry the
// same 16 points, supplying different K-rows of the 4x16 B matrix):
//   B vgpr0: lanes0-15 = K0 row (fx), lanes16-31 = K2 row (fz)
//   B vgpr1: lanes0-15 = K1 row (fy), lanes16-31 = K3 row (1.0)
// D rows 0..2 (lanes 0-15, vgprs 0..2) hold the transformed coords.
__global__ void geom_index(const float* __restrict__ frustum,
                           const float* __restrict__ wsMat,
                           int* __restrict__ wsIdx) {
    int lane = threadIdx.x & 31;
    int g = blockIdx.x * (blockDim.x >> 5) + (threadIdx.x >> 5);
    if (g >= NGROUPS) return;                 // wave-uniform: EXEC stays full
    int Lr = lane & 15;
    int pt = g * 16 + Lr;                     // 16-pt groups never cross a (b,n)
    int bn  = pt / PTS_PER_BN;
    int rem = pt % PTS_PER_BN;
    int dz = rem / (FHH*FWW);
    int r2 = rem % (FHH*FWW);
    int fh = r2 / FWW;
    int fw = r2 % FWW;
    const float* f = frustum + ((dz*FHH + fh)*FWW + fw)*3;
    float fx = f[0], fy = f[1], fz = f[2];

    const float* mbase = wsMat + bn*128;      // bn uniform across the wave
    v2f A, B;
    A.x = mbase[lane];
    A.y = mbase[32 + lane];
    B.x = (lane < 16) ? fx : fz;
    B.y = (lane < 16) ? fy : 1.0f;
    v8f c = {0.f,0.f,0.f,0.f,0.f,0.f,0.f,0.f};
    // q = invPR * (f - post_trans)
    v8f q = __builtin_amdgcn_wmma_f32_16x16x4_f32(false, A, false, B,
                                                  (short)0, c, false, false);
    float qx = q[0], qy = q[1], qz = q[2];    // valid in lanes 0-15
    float qzAll = __shfl(qz, Lr, 32);         // broadcast qz to upper half

    v2f A2, B2;
    A2.x = mbase[64 + lane];
    A2.y = mbase[96 + lane];
    B2.x = (lane < 16) ? qx*qz : qzAll;       // u = (qx*qz, qy*qz, qz)
    B2.y = (lane < 16) ? qy*qz : 1.0f;
    // r = (rot*invK) * u + trans
    v8f r = __builtin_amdgcn_wmma_f32_16x16x4_f32(false, A2, false, B2,
                                                  (short)0, c, false, false);
    float rx = r[0], ry = r[1], rz = r[2];

    // ((geom - (bx - dx/2)) / dx).astype(int32): trunc-toward-zero, exact consts
    int gx = (int)((rx + 50.0f) / 0.5f);
    int gy = (int)((ry + 50.0f) / 0.5f);
    int gz = (int)((rz + 10.0f) / 20.0f);
    bool kept = (gx >= 0) & (gx < NXV) & (gy >= 0) & (gy < NYV)
              & (gz >= 0) & (gz < NZV);
    int b = bn / NN;
    int base = kept ? (b*(CC*PLANE) + gy*NXV + gx) : -1;   // NZ==1 -> z folds out
    if (lane < 16) wsIdx[pt] = base;
}

__global__ void zero_out(float4* __restrict__ out, int n4) {
    int i = blockIdx.x * blockDim.x + threadIdx.x;
    if (i < n4) out[i] = make_float4(0.f, 0.f, 0.f, 0.f);
}

// Kernel 4: scatter-add. One thread = 4 consecutive channels of one point.
// Feature reads fully coalesced (16B) and NON-TEMPORAL: the 443 MB feature
// stream is read exactly once, while the 102 MB output must stay resident in
// the 192 MB L2 so the f32 atomics resolve at L2 rate. NT loads keep the
// stream from evicting the output. wsIdx (reused 20x/point) stays cached.
__global__ void scatter_add(const v4f* __restrict__ feat4,
                            const int* __restrict__ wsIdx,
                            float* __restrict__ out) {
    int tid = blockIdx.x * blockDim.x + threadIdx.x;
    const int TOT = NPTS * (CC/4);
    if (tid >= TOT) return;
    int p  = tid / (CC/4);
    int cg = tid % (CC/4);
    int t = wsIdx[p];
    if (t < 0) return;
    v4f v = __builtin_nontemporal_load(feat4 + tid);   // == feat[p*80 + cg*4 ..]
    float* o = out + t + cg*4*PLANE;
    unsafeAtomicAdd(o + 0*PLANE, v.x);
    unsafeAtomicAdd(o + 1*PLANE, v.y);
    unsafeAtomicAdd(o + 2*PLANE, v.z);
    unsafeAtomicAdd(o + 3*PLANE, v.w);
}

extern "C" void kernel_launch(void* const* d_in, const int* in_sizes, int n_in,
                              void* d_out, int out_size, void* d_ws, size_t ws_size,
                              hipStream_t stream) {
    const float* cam        = (const float*)d_in[0];
    const float* rots       = (const float*)d_in[1];
    const float* trans      = (const float*)d_in[2];
    const float* intrins    = (const float*)d_in[3];
    const float* post_rots  = (const float*)d_in[4];
    const float* post_trans = (const float*)d_in[5];
    const float* frustum    = (const float*)d_in[6];
    float* out = (float*)d_out;

    float* wsMat = (float*)d_ws;                        // 48*128 floats = 24 KB
    int*   wsIdx = (int*)((char*)d_ws + 32768);         // NPTS ints ~= 5.5 MB

    setup_mats<<<1, 64, 0, stream>>>(rots, trans, intrins, post_rots, post_trans, wsMat);

    geom_index<<<NGROUPS/8, 256, 0, stream>>>(frustum, wsMat, wsIdx);

    int n4 = out_size / 4;                              // 25.6M floats -> 6.4M float4
    zero_out<<<(n4 + 255)/256, 256, 0, stream>>>((float4*)out, n4);

    int tot = NPTS * (CC/4);
    scatter_add<<<(tot + 255)/256, 256, 0, stream>>>((const v4f*)cam, wsIdx, out);
}